// OuterProduct_53996328845341
// MI455X (gfx1250) — compile-verified
//
#include <hip/hip_runtime.h>

typedef __attribute__((ext_vector_type(2))) float v2f;
typedef __attribute__((ext_vector_type(8))) float v8f;

#define F_    32
#define D_    64
#define P_    496
#define NB    4096
#define ROWS  (F_ * D_)    // 2048 floats per batch row of x
#define OUTS  (F_ + P_)    // 528 floats per output row
#define WAVES 2
#define TPB   (WAVES * 32) // 64 threads per block
#define MT    2            // M-tiles (of 16 batches) per wave -> 32 batches/wave
#define XJPAD 35           // gcd(35,64)=1; cross-half collision at delta=24 (out of range)

// ---------------------------------------------------------------------------
// Kernel 1: z[b,f] = sum_d x[b,f,d]   (memory-bound, coalesced)
// ---------------------------------------------------------------------------
__global__ __launch_bounds__(256) void rowsum_kernel(
    const float* __restrict__ x, float* __restrict__ out)
{
  const int gt  = blockIdx.x * 256 + threadIdx.x;
  const int row = gt >> 4;            // b*F_ + f
  const int t   = gt & 15;
  const float4 v = *(const float4*)(x + (size_t)row * D_ + t * 4);
  float s = (v.x + v.y) + (v.z + v.w);
#pragma unroll
  for (int mask = 1; mask <= 8; mask <<= 1)
    s += __shfl_xor(s, mask, 32);
  if (t == 0) {
    const int b = row >> 5;
    const int f = row & (F_ - 1);
    out[(size_t)b * OUTS + f] = s;
  }
}

// ---------------------------------------------------------------------------
// Kernel 2: outer[b,p] = xi^T K_p xj  via V_WMMA_F32_16X16X4_F32
//   block = (pair p, 64-batch slab); each wave owns 32 batches (2 M-tiles).
//   Register blocking: B-fragments of K_p are reused across both M-tiles,
//   so 128 WMMAs share one set of 64 B-fragment ds_load_b64s.
// Fragment conventions (32-bit, wave32), lane l = h*16 + t, h = l>=16:
//   A frag  (16x4):  A[q] = Xi[m = t, k = k0 + 2h + q]
//   B frag  (4x16):  B[q] = K_p[k = k0 + 2h + q, n = n0 + t]
//   C frag (16x16):  C[r][l] = T[m = h*8 + r, n = n0 + t]
// ---------------------------------------------------------------------------
__global__ __launch_bounds__(TPB) void bilinear_wmma_kernel(
    const float* __restrict__ x, const float* __restrict__ kern,
    float* __restrict__ out)
{
  __shared__ float ldsK [16][4][64];            // [kc][nt][2*lane+q]  B-frag order
  __shared__ float ldsXi[WAVES][16][MT * 64];   // [wave][kc][mt*64 + 2*lane+q]
  __shared__ float ldsXj[WAVES][64][XJPAD];     // [wave][d][m], m = 0..31

  const int tid  = threadIdx.x;
  const int lane = tid & 31;
  const int wave = tid >> 5;
  const int t    = lane & 15;
  const int half = lane >> 4;
  const int p    = blockIdx.x;

  // unrank p -> (i, j) over triu(F_, k=1); scalar & wave-uniform
  int i = 0, rem = p;
  while (rem >= F_ - 1 - i) { rem -= F_ - 1 - i; ++i; }
  const int j = i + 1 + rem;

  const int bw = blockIdx.y * (WAVES * MT * 16) + wave * (MT * 16); // wave batch base

  // ---- stage K_p into LDS in B-fragment order (block cooperative) ----
  {
    const float* kp = kern + (size_t)p * (D_ * D_);
#pragma unroll
    for (int it = 0; it < (D_ * D_) / (TPB * 4); ++it) {  // 16 iters
      const int gidx = it * (TPB * 4) + tid * 4;          // coalesced float4
      const float4 v = *(const float4*)(kp + gidx);
      const int d  = gidx >> 6;
      const int e0 = gidx & 63;
      const int kc = d >> 2, h = (d >> 1) & 1, q = d & 1;
      const int nt = e0 >> 4;
      const float vv[4] = {v.x, v.y, v.z, v.w};
#pragma unroll
      for (int u = 0; u < 4; ++u)
        ldsK[kc][nt][2 * (h * 16 + (e0 & 15) + u) + q] = vv[u];
    }
  }

  // ---- stage xi (A-fragment order) + xj (transposed) for this wave ----
#pragma unroll
  for (int rep = 0; rep < 16; ++rep) {
    const int m  = rep * 2 + half;      // 2 batch rows per rep, 256B coalesced
    const int d0 = t * 4;
    const float4 vi = *(const float4*)(x + (size_t)(bw + m) * ROWS + i * D_ + d0);
    const float4 vj = *(const float4*)(x + (size_t)(bw + m) * ROWS + j * D_ + d0);
    const float vvi[4] = {vi.x, vi.y, vi.z, vi.w};
    const float vvj[4] = {vj.x, vj.y, vj.z, vj.w};
    const int mt = m >> 4, ml = m & 15;
#pragma unroll
    for (int u = 0; u < 4; ++u) {
      const int dd = d0 + u;
      const int kc = dd >> 2, h = (dd >> 1) & 1, q = dd & 1;
      ldsXi[wave][kc][mt * 64 + 2 * (h * 16 + ml) + q] = vvi[u];
      ldsXj[wave][dd][m] = vvj[u];
    }
  }
  __syncthreads();

  // ---- T = Xi @ K_p : 128 chained WMMAs, B-fragments shared across M-tiles ----
  v8f acc[MT][4] = {};   // 64 VGPRs of f32 accumulators
#pragma unroll
  for (int kc = 0; kc < 16; ++kc) {
    v2f a[MT];
#pragma unroll
    for (int mt = 0; mt < MT; ++mt)
      a[mt] = *(const v2f*)&ldsXi[wave][kc][mt * 64 + 2 * lane];  // ds_load_b64
#pragma unroll
    for (int nt = 0; nt < 4; ++nt) {
      const v2f b = *(const v2f*)&ldsK[kc][nt][2 * lane];         // ds_load_b64
#pragma unroll
      for (int mt = 0; mt < MT; ++mt)
        acc[mt][nt] = __builtin_amdgcn_wmma_f32_16x16x4_f32(
            false, a[mt], false, b, (short)0, acc[mt][nt], false, false);
    }
  }

  // ---- epilogue per M-tile: y[m] = sum_e T[m,e] * xj[m,e] ----
#pragma unroll
  for (int mt = 0; mt < MT; ++mt) {
    float partial[8];
#pragma unroll
    for (int r = 0; r < 8; ++r) {
      float s = 0.f;
#pragma unroll
      for (int nt = 0; nt < 4; ++nt)
        s += acc[mt][nt][r] * ldsXj[wave][nt * 16 + t][mt * 16 + half * 8 + r];
      partial[r] = s;
    }
#pragma unroll
    for (int mask = 1; mask <= 8; mask <<= 1)
#pragma unroll
      for (int r = 0; r < 8; ++r)
        partial[r] += __shfl_xor(partial[r], mask, 32);  // stays within half

    float yv = 0.f;
#pragma unroll
    for (int r = 0; r < 8; ++r) {
      const float hi = __shfl(partial[r], 16 + r, 32);   // half-1 sum
      if (lane == r)     yv = partial[r];
      if (lane == 8 + r) yv = hi;
    }
    if (lane < 16)
      out[(size_t)(bw + mt * 16 + lane) * OUTS + F_ + p] = yv;
  }
}

// ---------------------------------------------------------------------------
extern "C" void kernel_launch(void* const* d_in, const int* in_sizes, int n_in,
                              void* d_out, int out_size, void* d_ws, size_t ws_size,
                              hipStream_t stream) {
  const float* x    = (const float*)d_in[0];   // (4096, 32, 64) f32
  const float* kern = (const float*)d_in[1];   // (496, 64, 64)  f32
  float* out = (float*)d_out;                  // (4096, 528)    f32

  rowsum_kernel<<<dim3((NB * F_ * 16) / 256), 256, 0, stream>>>(x, out);
  bilinear_wmma_kernel<<<dim3(P_, NB / (WAVES * MT * 16)), TPB, 0, stream>>>(x, kern, out);
}